// LearnableGate_62551903699065
// MI455X (gfx1250) — compile-verified
//
#include <hip/hip_runtime.h>

// CDNA5 / gfx1250: wave32, WMMA 16x16x4 F32 (full-fp32 matrix pipe).
typedef __attribute__((ext_vector_type(2))) float v2f;
typedef __attribute__((ext_vector_type(8))) float v8f;

static constexpr int kD0   = 128;           // scalar block width
static constexpr int kD1   = 192;           // 64 * 3
static constexpr int kD2   = 160;           // 32 * 5
static constexpr int kDIN  = 480;           // total row width (in == out)
static constexpr int kLDJ0 = 240;           // W0 pair-row stride (2*240 % 64 == 32 -> bank tiling)
static constexpr int kLDJ1 = 80;            // W1 pair-row stride
static constexpr int kLDJ2 = 48;            // W2 pair-row stride

static constexpr float kS0 = 0.08838834764831845f;   // 1/sqrt(128)
static constexpr float kS1 = 0.125f;                  // 1/sqrt(64)
static constexpr float kS2 = 0.17677669529663689f;   // 1/sqrt(32)
static constexpr float kTN = 1.5925466f;              // TANH_NORM = 1/sqrt(E[tanh(z)^2])

__device__ __forceinline__ float fast_tanh(float v) {
    // tanh(v) = (e^{2v}-1) / (e^{2v}+1); clamp so e stays finite.
    v = fminf(fmaxf(v, -15.0f), 15.0f);
    float e = __expf(2.0f * v);                      // v_exp_f32
    return (e - 1.0f) * __builtin_amdgcn_rcpf(e + 1.0f);
}

__device__ __forceinline__ v8f wmma4(v2f a, v2f b, v8f c) {
    // D(16x16,f32) += A(16x4,f32) x B(4x16,f32)
    return __builtin_amdgcn_wmma_f32_16x16x4_f32(false, a, false, b, (short)0, c,
                                                 false, false);
}

__global__ __launch_bounds__(256)
void lg_wmma_kernel(const float* __restrict__ x,
                    const float* __restrict__ W0,
                    const float* __restrict__ W1,
                    const float* __restrict__ W2,
                    float* __restrict__ out,
                    int n_tiles)
{
    // Weights in LDS, row-PAIR interleaved: element [p*ld + j] = (W[2p][j], W[2p+1][j]).
    // One b64 LDS read yields the two K-values each lane-half needs; strides chosen so
    // the two 16-lane access groups tile all 64 banks (conflict-free).
    __shared__ v2f sW0[64 * kLDJ0];   // 122880 B
    __shared__ v2f sW1[32 * kLDJ1];   //  20480 B
    __shared__ v2f sW2[16 * kLDJ2];   //   6144 B   (total ~146 KB <= 320 KB/WGP)

    const int tid = threadIdx.x;
    for (int idx = tid; idx < 128 * 224; idx += 256) {
        int r = idx / 224, c = idx - r * 224;
        reinterpret_cast<float*>(&sW0[(r >> 1) * kLDJ0 + c])[r & 1] = W0[idx];
    }
    for (int idx = tid; idx < 64 * 64; idx += 256) {
        int r = idx >> 6, c = idx & 63;
        reinterpret_cast<float*>(&sW1[(r >> 1) * kLDJ1 + c])[r & 1] = W1[idx];
    }
    for (int idx = tid; idx < 32 * 32; idx += 256) {
        int r = idx >> 5, c = idx & 31;
        reinterpret_cast<float*>(&sW2[(r >> 1) * kLDJ2 + c])[r & 1] = W2[idx];
    }
    __syncthreads();

    const int lane = tid & 31;
    const int wv   = tid >> 5;
    const int m    = lane & 15;   // A row / B col / C col within 16-tile
    const int h    = lane >> 4;   // lane half: selects K offset (+2) for A/B frags

    const v8f zero8 = {0.f, 0.f, 0.f, 0.f, 0.f, 0.f, 0.f, 0.f};

    for (int tile = blockIdx.x * 8 + wv; tile < n_tiles; tile += gridDim.x * 8) {
        const float* xrow = x   + (size_t)(tile * 16 + m) * kDIN;  // this lane's A row
        float*       orow = out + (size_t)(tile * 16) * kDIN;

        __builtin_prefetch(xrow + kD0, 0, 0);        // gate/vector input regions
        __builtin_prefetch(xrow + kD0 + kD1, 0, 0);

        // ---- phase 0: full y0 = x0 @ W0 (cols 0..223) in one K sweep.
        // acc[0..7]  -> scalar tiles (cols 0..127)
        // acc[8..13] -> gate tiles   (cols 128..223)
        v8f acc[14];
        #pragma unroll
        for (int t = 0; t < 14; ++t) acc[t] = zero8;
        #pragma unroll 2
        for (int kk = 0; kk < 32; ++kk) {            // K = 128, 4 per WMMA
            v2f a = *reinterpret_cast<const v2f*>(xrow + 4 * kk + 2 * h);
            const v2f* wp = &sW0[(2 * kk + h) * kLDJ0 + m];
            #pragma unroll
            for (int t = 0; t < 14; ++t)
                acc[t] = wmma4(a, wp[16 * t], acc[t]);
        }

        // gates (register-resident; same C-tile layout as y1/y2 accumulators)
        float gate[6][8];
        #pragma unroll
        for (int t = 0; t < 6; ++t)
            #pragma unroll
            for (int r = 0; r < 8; ++r)
                gate[t][r] = kTN * fast_tanh(acc[8 + t][r] * kS0);

        // scalars -> out[:, 0:128]
        #pragma unroll
        for (int t = 0; t < 8; ++t)
            #pragma unroll
            for (int r = 0; r < 8; ++r)
                orow[(size_t)(r + 8 * h) * kDIN + 16 * t + m] =
                    kTN * fast_tanh(acc[t][r] * kS0);

        // ---- phase 1: y1 (3 channels, K = 64, cols stride-3 in x), gated by tiles 0..3
        for (int c = 0; c < 3; ++c) {
            v8f yacc[4];
            #pragma unroll
            for (int t = 0; t < 4; ++t) yacc[t] = zero8;
            #pragma unroll 4
            for (int kk = 0; kk < 16; ++kk) {
                const float* xp = xrow + kD0 + 3 * (4 * kk + 2 * h) + c;
                v2f a; a.x = xp[0]; a.y = xp[3];
                const v2f* wp = &sW1[(2 * kk + h) * kLDJ1 + m];
                #pragma unroll
                for (int t = 0; t < 4; ++t)
                    yacc[t] = wmma4(a, wp[16 * t], yacc[t]);
            }
            #pragma unroll
            for (int t = 0; t < 4; ++t)
                #pragma unroll
                for (int r = 0; r < 8; ++r)
                    orow[(size_t)(r + 8 * h) * kDIN + kD0 + 3 * (16 * t + m) + c] =
                        yacc[t][r] * kS1 * gate[t][r];
        }

        // ---- phase 2: y2 (5 channels, K = 32, cols stride-5 in x), gated by tiles 4..5
        for (int c = 0; c < 5; ++c) {
            v8f yacc[2];
            #pragma unroll
            for (int t = 0; t < 2; ++t) yacc[t] = zero8;
            #pragma unroll 4
            for (int kk = 0; kk < 8; ++kk) {
                const float* xp = xrow + kD0 + kD1 + 5 * (4 * kk + 2 * h) + c;
                v2f a; a.x = xp[0]; a.y = xp[5];
                const v2f* wp = &sW2[(2 * kk + h) * kLDJ2 + m];
                #pragma unroll
                for (int t = 0; t < 2; ++t)
                    yacc[t] = wmma4(a, wp[16 * t], yacc[t]);
            }
            #pragma unroll
            for (int t = 0; t < 2; ++t)
                #pragma unroll
                for (int r = 0; r < 8; ++r)
                    orow[(size_t)(r + 8 * h) * kDIN + kD0 + kD1 + 5 * (16 * t + m) + c] =
                        yacc[t][r] * kS2 * gate[4 + t][r];
        }
    }
}

extern "C" void kernel_launch(void* const* d_in, const int* in_sizes, int n_in,
                              void* d_out, int out_size, void* d_ws, size_t ws_size,
                              hipStream_t stream) {
    const float* x  = (const float*)d_in[0];
    const float* W0 = (const float*)d_in[1];
    const float* W1 = (const float*)d_in[2];
    const float* W2 = (const float*)d_in[3];
    float* out = (float*)d_out;

    const int n       = in_sizes[0] / kDIN;   // rows
    const int n_tiles = n / 16;               // 16 rows per wave-tile

    int blocks = 512;                         // 8 waves/block, grid-stride over tiles
    if (blocks * 8 > n_tiles) blocks = (n_tiles + 7) / 8;
    if (blocks < 1) blocks = 1;

    lg_wmma_kernel<<<dim3(blocks), dim3(256), 0, stream>>>(x, W0, W1, W2, out, n_tiles);
}